// MoE_7206955123114
// MI455X (gfx1250) — compile-verified
//
#include <hip/hip_runtime.h>
#include <math.h>

typedef float v2f __attribute__((ext_vector_type(2)));
typedef float v8f __attribute__((ext_vector_type(8)));
typedef unsigned int u32x4 __attribute__((ext_vector_type(4)));
typedef int i32x4 __attribute__((ext_vector_type(4)));
typedef int i32x8 __attribute__((ext_vector_type(8)));

#define NTOK 512
#define DDIM 512
#define HDIM 32
#define NEXP 32
#define MAXT 1024     // worst case: all 1024 assignments to one expert
#define XSTRIDE 520   // 512 + 2 pads of 4 DWORDs (TDM pad_interval=256dw, amount=4dw)
#define APAD 34       // conflict-free act reads: 17*(l1-l2) % 32 != 0

#if __has_builtin(__builtin_amdgcn_tensor_load_to_lds)
#define USE_TDM 1
#else
#define USE_TDM 0
#endif

static __device__ __forceinline__ v8f wmma4(v2f a, v2f b, v8f c) {
  // V_WMMA_F32_16X16X4_F32 : D = A(16x4) * B(4x16) + C, fp32 end-to-end
  return __builtin_amdgcn_wmma_f32_16x16x4_f32(false, a, false, b, (short)0, c,
                                               false, false);
}

static __device__ __forceinline__ float gelu_tanh(float v) {
  return 0.5f * v * (1.0f + tanhf(0.7978845608028654f * (v + 0.044715f * v * v * v)));
}

// ---------------- kernel 0: zero per-expert counters ----------------
__global__ __launch_bounds__(32) void moe_init(int* __restrict__ counts) {
  counts[threadIdx.x] = 0;
}

// ---------------- kernel 1: RMSNorm + router + top-2 + bucketize ----------------
__global__ __launch_bounds__(256) void moe_router(
    const float* __restrict__ x, const float* __restrict__ rscale,
    const float* __restrict__ rw, const float* __restrict__ pescale,
    int* __restrict__ counts, int* __restrict__ etok, float* __restrict__ ec) {
  __shared__ float ri[DDIM];
  __shared__ float red[256];
  __shared__ float part[8][NEXP];
  __shared__ float logits[NEXP];

  const int tok = blockIdx.x;
  const int tid = threadIdx.x;
  const float* xr = x + (size_t)tok * DDIM;

  const float v0 = xr[tid];
  const float v1 = xr[tid + 256];
  red[tid] = v0 * v0 + v1 * v1;
  __syncthreads();
  for (int off = 128; off > 0; off >>= 1) {
    if (tid < off) red[tid] += red[tid + off];
    __syncthreads();
  }
  const float var = red[0] * (1.0f / (float)DDIM);
  const float inv = rsqrtf(var + 1e-6f) * rsqrtf((float)DDIM);
  ri[tid]       = v0 * inv * rscale[tid];
  ri[tid + 256] = v1 * inv * rscale[tid + 256];
  __syncthreads();

  // logits[e] = sum_d ri[d] * rw[d][e];  thread t: expert t%32, d-chunk t/32
  const int e  = tid & 31;
  const int ch = tid >> 5;
  float p = 0.0f;
  const int dbeg = ch * 64;
#pragma unroll 8
  for (int d = dbeg; d < dbeg + 64; ++d) p += ri[d] * rw[(size_t)d * NEXP + e];
  part[ch][e] = p;
  __syncthreads();
  if (tid < NEXP) {
    float l = 0.0f;
#pragma unroll
    for (int c = 0; c < 8; ++c) l += part[c][tid];
    logits[tid] = l;
  }
  __syncthreads();

  if (tid == 0) {
    float m = -3.4e38f;
    for (int i = 0; i < NEXP; ++i) m = fmaxf(m, logits[i]);
    float pr[NEXP];
    float sum = 0.0f;
    for (int i = 0; i < NEXP; ++i) { pr[i] = expf(logits[i] - m); sum += pr[i]; }
    int i0 = 0;
    for (int i = 1; i < NEXP; ++i) if (logits[i] > logits[i0]) i0 = i;
    int i1 = (i0 == 0) ? 1 : 0;
    for (int i = 0; i < NEXP; ++i)
      if (i != i0 && logits[i] > logits[i1]) i1 = i;
    const float p0 = pr[i0] / sum, p1 = pr[i1] / sum;
    float rn = p0 + p1;
    rn = (rn > 0.0f) ? rn : 1.0f;
    const float c0 = pescale[i0] * (p0 / rn);
    const float c1 = pescale[i1] * (p1 / rn);
    int s0 = atomicAdd(&counts[i0], 1);
    etok[i0 * MAXT + s0] = tok * 2 + 0;
    ec[i0 * MAXT + s0]   = c0;
    int s1 = atomicAdd(&counts[i1], 1);
    etok[i1 * MAXT + s1] = tok * 2 + 1;
    ec[i1 * MAXT + s1]   = c1;
  }
}

// ---------------- kernel 2: expert-grouped WMMA GEMMs ----------------
__global__ __launch_bounds__(128) void moe_expert(
    const float* __restrict__ x, const float* __restrict__ gating,
    const float* __restrict__ linw, const int* __restrict__ counts,
    const int* __restrict__ etok, const float* __restrict__ ec,
    float* __restrict__ contrib) {
  __shared__ float lds_x[16 * XSTRIDE];        // gathered token rows (TDM-padded)
  __shared__ float lds_gate[2][16][HDIM];      // gate0 / gate1 results
  __shared__ float lds_act[16][APAD];          // gelu(g0)*g1*scale
  __shared__ int   slots[16];
  __shared__ float cvals[16];

  const int e  = blockIdx.x;
  const int t0 = blockIdx.y * 16;
  const int n  = counts[e];
  if (t0 >= n) return;

  const int tid  = threadIdx.x;
  const int wid  = tid >> 5;          // 4 waves (wave32)
  const int lane = tid & 31;
  const int nl   = lane & 15;         // M (A) / N (B,D) index within tile
  const int ks   = (lane >> 4) * 2;   // K-pair base for A/B fragments
  const int mo   = (lane >> 4) * 8;   // C/D row offset

  const float* wl = linw + (size_t)e * HDIM * DDIM;

  // ---- tile metadata ----
  if (tid < 16) {
    const int s = t0 + tid;
    if (s < n) { slots[tid] = etok[e * MAXT + s]; cvals[tid] = ec[e * MAXT + s]; }
    else       { slots[tid] = -1;                 cvals[tid] = 0.0f; }
  }
  __syncthreads();

#if USE_TDM
  // ---- gather 16 token rows via the Tensor Data Mover (gather mode) ----
  if (wid == 0) {
    unsigned w[8];
#pragma unroll
    for (int i = 0; i < 8; ++i) {
      const int p0 = slots[2 * i], p1 = slots[2 * i + 1];
      const unsigned r0 = (p0 >= 0) ? (unsigned)(p0 >> 1) : 0u;
      const unsigned r1 = (p1 >= 0) ? (unsigned)(p1 >> 1) : 0u;
      w[i] = r0 | (r1 << 16);
    }
    const unsigned lds_base = (unsigned)(size_t)&lds_x[0];
    const unsigned long long ga = (unsigned long long)(size_t)x;
    // group0: count=1, gather_mode=1 (16-bit idx), lds_addr, global_addr, type=2
    u32x4 g0 = {0x80000001u, lds_base, (unsigned)ga,
                ((unsigned)(ga >> 32) & 0x01FFFFFFu) | 0x80000000u};
    // group1: data_size=4B, pad_enable, pad_interval=256dw, pad_amount=4dw,
    //         tensor_dim0=512, tensor_dim1=512, tile_dim0=512, tile_dim1=16 idx,
    //         tensor_dim0_stride=512
    i32x8 g1 = {0x07D20000, 0x02000000, 0x02000000, 0x02000000, 16, 512, 0, 0};
    i32x4 g2 = {(int)w[0], (int)w[1], (int)w[2], (int)w[3]};
    i32x4 g3 = {(int)w[4], (int)w[5], (int)w[6], (int)w[7]};
    i32x8 g4 = {0, 0, 0, 0, 0, 0, 0, 0};   // VADDR4 group: unused per ISA, zero
    __builtin_amdgcn_tensor_load_to_lds(g0, g1, g2, g3, g4, 0);
  }
  // overlap the DMA with prefetch of GEMM2 weights (64 KB -> global_prefetch_b8)
  for (int off = tid; off < (HDIM * DDIM * 4) / 128; off += 128)
    __builtin_prefetch((const char*)wl + (size_t)off * 128, 0, 3);
  if (wid == 0) __builtin_amdgcn_s_wait_tensorcnt(0);
#else
  for (int off = tid; off < (HDIM * DDIM * 4) / 128; off += 128)
    __builtin_prefetch((const char*)wl + (size_t)off * 128, 0, 3);
  for (int idx = tid; idx < 16 * (DDIM / 4); idx += 128) {
    const int row = idx >> 7;             // DDIM/4 = 128 float4 per row
    const int c4  = idx & 127;
    const int pk  = slots[row];
    const int tr  = (pk >= 0) ? (pk >> 1) : 0;
    const float4 v = *(const float4*)(x + (size_t)tr * DDIM + c4 * 4);
    *(float4*)&lds_x[row * XSTRIDE + c4 * 4 + ((c4 >= 64) ? 4 : 0)] = v;
  }
#endif
  __syncthreads();

  // ---- GEMM1: gate_g[t][h] = sum_d x[t][d] * W[e][g][h][d] ----
  // wave w -> gate g = w&1, h-tile base = (w>>1)*16 ; dual accumulators
  {
    const int g  = wid & 1;
    const int hb = (wid >> 1) * 16;
    const float* wg = gating + (((size_t)e * 2 + g) * HDIM + (hb + nl)) * DDIM;
    const float* a0 = lds_x + nl * XSTRIDE + ks;       // first half (d < 256)
    const float* a1 = a0 + 4;                          // second half (+pad skip)
    v8f acc0 = {}, acc1 = {};
#pragma unroll 4
    for (int d0 = 0; d0 < 256; d0 += 8) {
      acc0 = wmma4(*(const v2f*)(a0 + d0),     *(const v2f*)(wg + d0 + ks),     acc0);
      acc1 = wmma4(*(const v2f*)(a0 + d0 + 4), *(const v2f*)(wg + d0 + 4 + ks), acc1);
    }
#pragma unroll 4
    for (int d0 = 256; d0 < 512; d0 += 8) {
      acc0 = wmma4(*(const v2f*)(a1 + d0),     *(const v2f*)(wg + d0 + ks),     acc0);
      acc1 = wmma4(*(const v2f*)(a1 + d0 + 4), *(const v2f*)(wg + d0 + 4 + ks), acc1);
    }
    const v8f acc = acc0 + acc1;
#pragma unroll
    for (int r = 0; r < 8; ++r) lds_gate[g][mo + r][hb + nl] = acc[r];
  }
  __syncthreads();

  // ---- activation: act[t][h] = gelu(g0)*g1 * (escale*weight)[t] ----
  for (int idx = tid; idx < 16 * HDIM; idx += 128) {
    const int t = idx >> 5, h = idx & 31;
    const float g0 = lds_gate[0][t][h];
    const float g1 = lds_gate[1][t][h];
    lds_act[t][h] = gelu_tanh(g0) * g1 * cvals[t];
  }
  __syncthreads();

  // ---- GEMM2: out[t][d] = sum_h act[t][h] * linw[e][h][d] ----
  for (int dt = wid; dt < DDIM / 16; dt += 4) {
    const int db = dt * 16;
    v8f acc0 = {}, acc1 = {};
#pragma unroll
    for (int h0 = 0; h0 < HDIM; h0 += 8) {
      {
        const v2f a = *(const v2f*)&lds_act[nl][h0 + ks];
        v2f b;
        b.x = wl[(size_t)(h0 + ks)     * DDIM + db + nl];
        b.y = wl[(size_t)(h0 + ks + 1) * DDIM + db + nl];
        acc0 = wmma4(a, b, acc0);
      }
      {
        const v2f a = *(const v2f*)&lds_act[nl][h0 + 4 + ks];
        v2f b;
        b.x = wl[(size_t)(h0 + 4 + ks)     * DDIM + db + nl];
        b.y = wl[(size_t)(h0 + 4 + ks + 1) * DDIM + db + nl];
        acc1 = wmma4(a, b, acc1);
      }
    }
    const v8f acc = acc0 + acc1;
#pragma unroll
    for (int r = 0; r < 8; ++r) {
      const int pk = slots[mo + r];
      if (pk >= 0) contrib[(size_t)pk * DDIM + db + nl] = acc[r];
    }
  }
}

// ---------------- kernel 3: combine the two expert contributions ----------------
__global__ __launch_bounds__(256) void moe_combine(const float* __restrict__ contrib,
                                                   float* __restrict__ out) {
  const int i  = blockIdx.x * 256 + threadIdx.x;   // NTOK*DDIM threads
  const int t  = i >> 9;
  const int d  = i & (DDIM - 1);
  out[i] = contrib[(size_t)t * (2 * DDIM) + d] +
           contrib[(size_t)t * (2 * DDIM) + DDIM + d];
}

extern "C" void kernel_launch(void* const* d_in, const int* in_sizes, int n_in,
                              void* d_out, int out_size, void* d_ws, size_t ws_size,
                              hipStream_t stream) {
  (void)in_sizes; (void)n_in; (void)out_size; (void)ws_size;
  const float* x       = (const float*)d_in[0];
  const float* rscale  = (const float*)d_in[1];
  const float* rw      = (const float*)d_in[2];
  const float* gating  = (const float*)d_in[3];
  const float* linw    = (const float*)d_in[4];
  const float* pescale = (const float*)d_in[5];
  float* out = (float*)d_out;

  char* ws = (char*)d_ws;
  int*   counts  = (int*)ws;                                    // 128 B
  int*   etok    = (int*)(ws + 128);                            // 32*1024*4 = 128 KB
  float* ec      = (float*)(ws + 128 + NEXP * MAXT * 4);        // 128 KB
  float* contrib = (float*)(ws + 128 + 2 * NEXP * MAXT * 4);    // 1024*512*4 = 2 MB

  moe_init<<<1, 32, 0, stream>>>(counts);
  moe_router<<<NTOK, 256, 0, stream>>>(x, rscale, rw, pescale, counts, etok, ec);
  moe_expert<<<dim3(NEXP, MAXT / 16), 128, 0, stream>>>(x, gating, linw, counts,
                                                        etok, ec, contrib);
  moe_combine<<<(NTOK * DDIM) / 256, 256, 0, stream>>>(contrib, out);
}